// EvolveGCN_8899172237846
// MI455X (gfx1250) — compile-verified
//
#include <hip/hip_runtime.h>
#include <math.h>

#define NN   200000
#define NE   500000
#define F    166
#define FP   176          // F padded to 11*16
#define HH   128
#define TT   49
#define F3   498          // 3*F
#define ORDER_SLOTS (NN + TT*16)   // 200784, multiple of 16
#define NTILES (ORDER_SLOTS/16)    // 12549

typedef float v2f __attribute__((ext_vector_type(2)));
typedef float v8f __attribute__((ext_vector_type(8)));

// ---------------- utility kernels ----------------
__global__ void zero_f32(float* p, long n) {
  long i = (long)blockIdx.x * blockDim.x + threadIdx.x;
  long s = (long)gridDim.x * blockDim.x;
  for (; i < n; i += s) p[i] = 0.0f;
}
__global__ void fill_i32(int* p, int v, long n) {
  long i = (long)blockIdx.x * blockDim.x + threadIdx.x;
  long s = (long)gridDim.x * blockDim.x;
  for (; i < n; i += s) p[i] = v;
}
// copy initial_w (F x F) into padded slot 0 of W_all (FP x FP)
__global__ void init_W0(const float* __restrict__ iw, float* __restrict__ W0) {
  int i = blockIdx.x * blockDim.x + threadIdx.x;
  if (i >= FP * FP) return;
  int r = i / FP, c = i % FP;
  W0[i] = (r < F && c < F) ? iw[r * F + c] : 0.0f;
}
// proj_w [H,F] -> projT [FP,H] (zero-padded rows)
__global__ void transpose_proj(const float* __restrict__ pw, float* __restrict__ pT) {
  int i = blockIdx.x * blockDim.x + threadIdx.x;
  if (i >= FP * HH) return;
  int f = i / HH, h = i % HH;
  pT[i] = (f < F) ? pw[h * F + f] : 0.0f;
}

// ---------------- GRU weight evolution (one step) ----------------
__global__ __launch_bounds__(256)
void gru_step(const float* __restrict__ Wt, float* __restrict__ Wt1,
              const float* __restrict__ wih, const float* __restrict__ whh,
              const float* __restrict__ bih, const float* __restrict__ bhh) {
  int r = blockIdx.x;               // row 0..F-1
  int tid = threadIdx.x;
  __shared__ float wrow[F];
  __shared__ float gi[F3];
  __shared__ float gh[F3];
  if (tid < F) wrow[tid] = Wt[r * FP + tid];
  __syncthreads();
  for (int j = tid; j < F3; j += 256) {
    float s1 = 0.0f, s2 = 0.0f;
    const float* a = wih + (size_t)j * F;
    const float* b = whh + (size_t)j * F;
    for (int k = 0; k < F; ++k) { float w = wrow[k]; s1 += w * a[k]; s2 += w * b[k]; }
    gi[j] = s1 + bih[j];
    gh[j] = s2 + bhh[j];
  }
  __syncthreads();
  for (int c = tid; c < FP; c += 256) {
    float out = 0.0f;
    if (c < F) {
      float rg = 1.0f / (1.0f + expf(-(gi[c] + gh[c])));
      float zg = 1.0f / (1.0f + expf(-(gi[F + c] + gh[F + c])));
      float ng = tanhf(gi[2 * F + c] + rg * gh[2 * F + c]);
      out = (1.0f - zg) * ng + zg * wrow[c];
    }
    Wt1[r * FP + c] = out;
  }
}

// ---------------- degree table ----------------
__global__ void count_edges(const int* __restrict__ src, const int* __restrict__ dst,
                            const int* __restrict__ ts, float* __restrict__ tbl) {
  long e = (long)blockIdx.x * blockDim.x + threadIdx.x;
  long s = (long)gridDim.x * blockDim.x;
  for (; e < NE; e += s) {
    int a = ts[src[e]], b = ts[dst[e]];
    int t = a > b ? a : b;
    unsafeAtomicAdd(&tbl[(size_t)dst[e] * TT + t], 1.0f);
  }
}
__global__ void prefix_rsqrt(float* __restrict__ tbl) {
  long i = (long)blockIdx.x * blockDim.x + threadIdx.x;
  long s = (long)gridDim.x * blockDim.x;
  for (; i < NN; i += s) {
    float deg = 1.0f;
    float* row = tbl + i * TT;
    for (int t = 0; t < TT; ++t) { deg += row[t]; row[t] = rsqrtf(deg); }
  }
}

// ---------------- x aggregation ----------------
__global__ void init_agg(const float* __restrict__ x, const int* __restrict__ ts,
                         const float* __restrict__ tbl, float* __restrict__ agg) {
  long i = (long)blockIdx.x * blockDim.x + threadIdx.x;
  long s = (long)gridDim.x * blockDim.x;
  const long n = (long)NN * FP;
  for (; i < n; i += s) {
    long node = i / FP; int f = (int)(i % FP);
    float v = 0.0f;
    if (f < F) {
      float dv = tbl[node * TT + ts[node]];
      v = x[node * F + f] * dv * dv;   // self-loop term: h/deg
    }
    agg[i] = v;
  }
}
__global__ __launch_bounds__(256)
void edge_agg(const int* __restrict__ src, const int* __restrict__ dst,
              const int* __restrict__ ts, const float* __restrict__ tbl,
              const float* __restrict__ x, float* __restrict__ agg) {
  int w = blockIdx.x * (blockDim.x >> 5) + (threadIdx.x >> 5);
  int lane = threadIdx.x & 31;
  if (w >= NE) return;
  int s = src[w], d = dst[w];
  int tsv = ts[s], td = ts[d];
  if (tsv > td) return;                          // edge never active at dst's output time
  float nw = tbl[(size_t)s * TT + td] * tbl[(size_t)d * TT + td];
  const float* xs = x + (size_t)s * F;
  float* ad = agg + (size_t)d * FP;
  for (int f = lane; f < F; f += 32)
    unsafeAtomicAdd(&ad[f], nw * xs[f]);
}

// ---------------- node ordering (counting sort by time step) ----------------
__global__ void hist_ts(const int* __restrict__ ts, int* __restrict__ bins) {
  long i = (long)blockIdx.x * blockDim.x + threadIdx.x;
  long s = (long)gridDim.x * blockDim.x;
  for (; i < NN; i += s) atomicAdd(&bins[ts[i]], 1);
}
__global__ void scan_bins(const int* __restrict__ bins, int* __restrict__ cursors) {
  int off = 0;
  for (int t = 0; t < TT; ++t) {
    cursors[t] = off;
    off += ((bins[t] + 15) / 16) * 16;     // 16-aligned group regions
  }
}
__global__ void scatter_nodes(const int* __restrict__ ts, int* __restrict__ cursors,
                              int* __restrict__ order) {
  long i = (long)blockIdx.x * blockDim.x + threadIdx.x;
  long s = (long)gridDim.x * blockDim.x;
  for (; i < NN; i += s) {
    int p = atomicAdd(&cursors[ts[i]], 1);
    order[p] = (int)i;
  }
}

// ---------------- fused per-node pipeline: agg@W_t -> relu(proj) -> cls ----------------
__global__ __launch_bounds__(64)
void node_pipeline(const int* __restrict__ order, const int* __restrict__ ts,
                   const float* __restrict__ agg, const float* __restrict__ Wall,
                   const float* __restrict__ pT, const float* __restrict__ pb,
                   const float* __restrict__ cw, const float* __restrict__ cb,
                   float* __restrict__ out) {
  __shared__ float smem[2 * 2 * 16 * FP];    // per wave: A tile + C1 tile
  const int wave = threadIdx.x >> 5;
  const int lane = threadIdx.x & 31;
  const int tile = blockIdx.x * 2 + wave;
  if (tile >= NTILES) return;
  float* A  = smem + wave * (2 * 16 * FP);
  float* C1 = A + 16 * FP;
  const int* base = order + tile * 16;
  const int id0 = base[0];
  if (id0 < 0) return;                       // empty padded tile (wave-uniform)
  const int t = ts[id0];
  const float* Wt = Wall + (size_t)(t + 1) * FP * FP;

  // stage A tile (16 nodes x FP) into LDS; padded rows -> 0
  for (int r = 0; r < 16; ++r) {
    int id = base[r];
    const float* srcp = agg + (size_t)(id < 0 ? 0 : id) * FP;
    for (int f = lane; f < FP; f += 32)
      A[r * FP + f] = (id >= 0) ? srcp[f] : 0.0f;
  }

  const int hl = lane >> 4;                  // half-wave (k split)
  const int ll = lane & 15;                  // m / n within fragment

  // GEMM1: C1[16xFP] = A[16xFP] x Wt[FPxFP]  (fp32 WMMA, K steps of 4)
  for (int nb = 0; nb < FP / 16; ++nb) {
    v8f acc = {};
    const int ncol = nb * 16 + ll;
    for (int ks = 0; ks < FP / 4; ++ks) {
      const int k = ks * 4 + hl * 2;
      v2f a = *(const v2f*)&A[ll * FP + k];
      v2f b;
      b.x = Wt[(size_t)(k + 0) * FP + ncol];
      b.y = Wt[(size_t)(k + 1) * FP + ncol];
      acc = __builtin_amdgcn_wmma_f32_16x16x4_f32(false, a, false, b,
                                                  (short)0, acc, false, false);
    }
    for (int v = 0; v < 8; ++v)
      C1[(v + 8 * hl) * FP + ncol] = acc[v];
  }

  // GEMM2: Z[16xH] = relu(C1 x projT + pb); Z overlays A (no longer needed)
  float* Z = A;
  for (int nb = 0; nb < HH / 16; ++nb) {
    v8f acc = {};
    const int ncol = nb * 16 + ll;
    for (int ks = 0; ks < FP / 4; ++ks) {
      const int k = ks * 4 + hl * 2;
      v2f a = *(const v2f*)&C1[ll * FP + k];
      v2f b;
      b.x = pT[(size_t)(k + 0) * HH + ncol];
      b.y = pT[(size_t)(k + 1) * HH + ncol];
      acc = __builtin_amdgcn_wmma_f32_16x16x4_f32(false, a, false, b,
                                                  (short)0, acc, false, false);
    }
    const float bias = pb[ncol];
    for (int v = 0; v < 8; ++v) {
      float z = acc[v] + bias;
      Z[(v + 8 * hl) * HH + ncol] = z > 0.0f ? z : 0.0f;
    }
  }

  // classifier: 2 outputs per node, lanes 0..15 own one row each
  if (lane < 16) {
    int id = base[lane];
    if (id >= 0) {
      float s0 = cb[0], s1 = cb[1];
      const float* zr = Z + lane * HH;
      for (int k = 0; k < HH; ++k) {
        float z = zr[k];
        s0 += z * cw[k];
        s1 += z * cw[HH + k];
      }
      out[(size_t)id * 2 + 0] = s0;
      out[(size_t)id * 2 + 1] = s1;
    }
  }
}

// ---------------- launch ----------------
extern "C" void kernel_launch(void* const* d_in, const int* in_sizes, int n_in,
                              void* d_out, int out_size, void* d_ws, size_t ws_size,
                              hipStream_t stream) {
  const float* x   = (const float*)d_in[0];
  const int*   ei  = (const int*)d_in[1];
  const int*   ts  = (const int*)d_in[2];
  const float* iw  = (const float*)d_in[3];
  const float* wih = (const float*)d_in[4];
  const float* whh = (const float*)d_in[5];
  const float* bih = (const float*)d_in[6];
  const float* bhh = (const float*)d_in[7];
  const float* pw  = (const float*)d_in[8];
  const float* pb  = (const float*)d_in[9];
  const float* cw  = (const float*)d_in[10];
  const float* cb  = (const float*)d_in[11];
  const int* src = ei;
  const int* dst = ei + NE;
  float* outp = (float*)d_out;

  float* ws = (float*)d_ws;
  size_t o = 0;
  float* Wall = ws + o;  o += (size_t)(TT + 1) * FP * FP;   // 50 padded FxF slots
  float* pT   = ws + o;  o += (size_t)FP * HH;
  float* tbl  = ws + o;  o += (size_t)NN * TT;              // dinv table
  float* agg  = ws + o;  o += (size_t)NN * FP;              // aggregated x
  int* order  = (int*)(ws + o); o += ORDER_SLOTS;
  int* bins   = (int*)(ws + o); o += TT;
  int* curs   = (int*)(ws + o); o += TT;

  const int ZB = 2048, ZT = 256;
  // init scratch
  zero_f32<<<ZB, ZT, 0, stream>>>(Wall, (long)(TT + 1) * FP * FP);
  zero_f32<<<ZB, ZT, 0, stream>>>(tbl, (long)NN * TT);
  fill_i32<<<64, ZT, 0, stream>>>(order, -1, (long)ORDER_SLOTS);
  fill_i32<<<1, 128, 0, stream>>>(bins, 0, 2L * TT);
  init_W0<<<(FP * FP + 255) / 256, 256, 0, stream>>>(iw, Wall);
  transpose_proj<<<(FP * HH + 255) / 256, 256, 0, stream>>>(pw, pT);

  // GRU weight chain (sequential)
  for (int t = 0; t < TT; ++t)
    gru_step<<<F, 256, 0, stream>>>(Wall + (size_t)t * FP * FP,
                                    Wall + (size_t)(t + 1) * FP * FP,
                                    wih, whh, bih, bhh);

  // degree / dinv table
  count_edges<<<1024, 256, 0, stream>>>(src, dst, ts, tbl);
  prefix_rsqrt<<<1024, 256, 0, stream>>>(tbl);

  // aggregate raw features
  init_agg<<<ZB, ZT, 0, stream>>>(x, ts, tbl, agg);
  edge_agg<<<(NE + 7) / 8, 256, 0, stream>>>(src, dst, ts, tbl, x, agg);

  // group nodes by time step
  hist_ts<<<1024, 256, 0, stream>>>(ts, bins);
  scan_bins<<<1, 1, 0, stream>>>(bins, curs);
  scatter_nodes<<<1024, 256, 0, stream>>>(ts, curs, order);

  // fused WMMA pipeline: agg @ W_t -> relu(proj) -> classifier
  node_pipeline<<<(NTILES + 1) / 2, 64, 0, stream>>>(order, ts, agg, Wall, pT,
                                                     pb, cw, cb, outp);
}